// ButterflyConv2dBBT_71012989272065
// MI455X (gfx1250) — compile-verified
//
#include <hip/hip_runtime.h>
#include <hip/hip_bf16.h>

// B=16, C=256, H=W=32, K=3, KK=9, N=256, rows R = 16384, GEMM K = 2304.
// out[j][r] = sum_k W[k][j] * im2col[k][r]  (M-dim = j, N-dim = r)

typedef __attribute__((ext_vector_type(2))) float v2f;
typedef __attribute__((ext_vector_type(8))) float v8f;

#define NCH   256
#define KKPAT 9
#define KDIM  (KKPAT * NCH)     // 2304
#define ROWS  16384
#define HW    1024
#define HP    34                // padded spatial extent
#define XPSZ  (16 * HP * HP * NCH)   // padded channel-last x, floats

// ---------------------------------------------------------------------------
// Prep: zero the padded channel-last image (borders stay zero => no halo masks)
__global__ __launch_bounds__(256) void bf_xpad_zero(float* __restrict__ xp) {
    size_t idx = (size_t)blockIdx.x * 256 + threadIdx.x;   // XPSZ total
    xp[idx] = 0.0f;
}

// Prep: interior fill xp[b][y+1][x+1][c] = x[b][c][y][x] via LDS-tiled transpose
// (coalesced on both sides).  grid = (16*32 rows, 8 channel groups), 256 thr.
__global__ __launch_bounds__(256) void bf_xpad_fill(const float* __restrict__ x,
                                                    float* __restrict__ xp) {
    __shared__ float t[32][33];
    const int by = blockIdx.x;             // b*32 + y
    const int cg = blockIdx.y;             // channel group: ch0 = cg*32
    const int b = by >> 5, y = by & 31;
    const int tid = threadIdx.x;
    const int xx = tid & 31, cc = tid >> 5;        // read: lanes sweep x
    const float* src = x + ((size_t)(b * NCH + cg * 32)) * HW + y * 32;
#pragma unroll
    for (int i = 0; i < 4; ++i)
        t[cc + 8 * i][xx] = src[(size_t)(cc + 8 * i) * HW + xx];
    __syncthreads();
    const int c = tid & 31, xw = tid >> 5;         // write: lanes sweep ch
    float* dst = xp + (((size_t)(b * HP + y + 1)) * HP + 1) * NCH + cg * 32 + c;
#pragma unroll
    for (int i = 0; i < 4; ++i)
        dst[(size_t)(xw + 8 * i) * NCH] = t[c][xw + 8 * i];
}

// Prep: M_s := identity (9 x 256 x 256)
__global__ __launch_bounds__(256) void bf_init_identity(float* __restrict__ M) {
    int idx = blockIdx.x * 256 + threadIdx.x;
    int c = idx & 255, r = (idx >> 8) & 255;
    M[idx] = (r == c) ? 1.0f : 0.0f;
}

// Prep: one butterfly stage (M <- T_st * M), twiddle (9,255,2,2)
__global__ __launch_bounds__(256) void bf_stage(const float* __restrict__ tw, int st,
                                                const float* __restrict__ Min,
                                                float* __restrict__ Mout) {
    int idx = blockIdx.x * 256 + threadIdx.x;      // 9*128*256
    int c  = idx & 255;
    int pl = (idx >> 8) & 127;
    int s  = idx >> 15;
    int g  = pl / st;
    int l  = pl - g * st;
    int r0 = g * (2 * st) + l;
    int r1 = r0 + st;
    const float* t = tw + (size_t)((s * 255) + (st - 1) + l) * 4;
    float t00 = t[0], t01 = t[1], t10 = t[2], t11 = t[3];
    int base = s << 16;
    float x0 = Min[base + (r0 << 8) + c];
    float x1 = Min[base + (r1 << 8) + c];
    Mout[base + (r0 << 8) + c] = t00 * x0 + t01 * x1;
    Mout[base + (r1 << 8) + c] = t10 * x0 + t11 * x1;
}

// Prep: packed weight Wp[k>>1][j][k&1] = M_s[j][c]/9  (k = s*256+c)
// => one A-fragment lane reads its two K-elements with a single b64 load.
__global__ __launch_bounds__(256) void bf_finalize(const float* __restrict__ M,
                                                   float* __restrict__ Wp) {
    int idx = blockIdx.x * 256 + threadIdx.x;      // KDIM*256
    int v  = idx & 1;
    int j  = (idx >> 1) & 255;
    int kh = idx >> 9;                             // 0..1151
    int k  = kh * 2 + v;
    int s  = k >> 8, c = k & 255;
    Wp[idx] = M[(s << 16) + (j << 8) + c] * (1.0f / 9.0f);
}

// Prep: bsum[j] = sum_s bias[s][j] / 9
__global__ __launch_bounds__(256) void bf_bias(const float* __restrict__ bias,
                                               float* __restrict__ bsum) {
    int j = threadIdx.x;
    float a = 0.0f;
#pragma unroll
    for (int s = 0; s < KKPAT; ++s) a += bias[s * NCH + j];
    bsum[j] = a * (1.0f / 9.0f);
}

// ---------------------------------------------------------------------------
// Main GEMM, V_WMMA_F32_16X16X4_F32.
// Block: 256 threads = 8 waves (2 M x 4 N).  Wave tile 32j x 64r.
// Block tile 64j x 256r.  Grid (ROWS/256, NCH/64).
// Inner K-quad: 2 b64 A-loads + 4 b64 B-loads + 8 WMMA, no masks, no VALU.
__global__ __launch_bounds__(256) void butterfly_gemm_wmma(
    const float* __restrict__ xp,    // (16,34,34,256) channel-last, zero-padded
    const float* __restrict__ Wp,    // (1152,256,2)
    const float* __restrict__ bsum,  // (256)
    float* __restrict__ out)         // (16,256,32,32)
{
    const int lane = threadIdx.x & 31;
    const int wid  = threadIdx.x >> 5;
    const int hi   = lane >> 4;
    const int ml   = lane & 15;

    const int r0 = blockIdx.x * 256;
    const int j0 = blockIdx.y * 64;
    const int jbase = j0 + (wid >> 2) * 32;
    const int rbase = r0 + (wid & 3) * 64;
    const int b = rbase >> 10;

    int ynf[4], x0nf[4];
#pragma unroll
    for (int nf = 0; nf < 4; ++nf) {
        int rn = rbase + 16 * nf;
        ynf[nf]  = (rn >> 5) & 31;
        x0nf[nf] = rn & 31;
    }

    const v8f vzero = {0.f, 0.f, 0.f, 0.f, 0.f, 0.f, 0.f, 0.f};
    v8f acc[2][4];
#pragma unroll
    for (int mf = 0; mf < 2; ++mf)
#pragma unroll
        for (int nf = 0; nf < 4; ++nf) acc[mf][nf] = vzero;

    for (int s = 0; s < KKPAT; ++s) {
        const int di = s / 3, dj = s % 3;
        // B base pointers: xp[b][y+di][x+dj][2*hi]  (padding absorbs the halo)
        const float* pB[4];
#pragma unroll
        for (int nf = 0; nf < 4; ++nf)
            pB[nf] = xp + (((size_t)(b * HP + ynf[nf] + di)) * HP
                           + (x0nf[nf] + ml + dj)) * NCH + 2 * hi;
        // A base pointer: Wp[(s*256)>>1 + hi][jbase + ml][0]
        const float* pA = Wp + ((size_t)(s * 128 + hi)) * 512 + (jbase + ml) * 2;

        for (int kq = 0; kq < 64; ++kq) {        // c0 = 4*kq
            v2f a0 = *(const v2f*)(pA + (size_t)kq * 1024);        // mf=0
            v2f a1 = *(const v2f*)(pA + (size_t)kq * 1024 + 32);   // mf=1
            v2f bf[4];
#pragma unroll
            for (int nf = 0; nf < 4; ++nf)
                bf[nf] = *(const v2f*)(pB[nf] + 4 * kq);
#pragma unroll
            for (int nf = 0; nf < 4; ++nf) {
                acc[0][nf] = __builtin_amdgcn_wmma_f32_16x16x4_f32(
                    false, a0, false, bf[nf], (short)0, acc[0][nf], false, false);
                acc[1][nf] = __builtin_amdgcn_wmma_f32_16x16x4_f32(
                    false, a1, false, bf[nf], (short)0, acc[1][nf], false, false);
            }
        }
    }

    // Store: j = jbase + 16*mf + q + 8*hi, r = rbase + 16*nf + ml
#pragma unroll
    for (int mf = 0; mf < 2; ++mf) {
        float bs[8];
#pragma unroll
        for (int q = 0; q < 8; ++q) bs[q] = bsum[jbase + 16 * mf + q + 8 * hi];
#pragma unroll
        for (int nf = 0; nf < 4; ++nf) {
            int r   = rbase + 16 * nf + ml;
            int pos = r & (HW - 1);
#pragma unroll
            for (int q = 0; q < 8; ++q) {
                int j = jbase + 16 * mf + q + 8 * hi;
                out[((size_t)b * NCH + j) * HW + pos] = acc[mf][nf][q] + bs[q];
            }
        }
    }
}

// ---------------------------------------------------------------------------
extern "C" void kernel_launch(void* const* d_in, const int* in_sizes, int n_in,
                              void* d_out, int out_size, void* d_ws, size_t ws_size,
                              hipStream_t stream) {
    const float* x    = (const float*)d_in[0];  // (16,256,32,32)
    const float* tw1  = (const float*)d_in[1];  // (9,255,2,2)
    const float* tw2  = (const float*)d_in[2];  // (9,255,2,2)
    const float* bias = (const float*)d_in[3];  // (9,256)
    float* out = (float*)d_out;

    // Workspace (floats): Wp | bufA | bufB | bsum | xp   (~26 MB total)
    float* Wp   = (float*)d_ws;
    float* bufA = Wp + (size_t)KDIM * NCH;
    float* bufB = bufA + (size_t)KKPAT * NCH * NCH;
    float* bsum = bufB + (size_t)KKPAT * NCH * NCH;
    float* xp   = bsum + 256;
    (void)ws_size; (void)in_sizes; (void)n_in; (void)out_size;

    // Padded channel-last image
    bf_xpad_zero<<<XPSZ / 256, 256, 0, stream>>>(xp);
    bf_xpad_fill<<<dim3(16 * 32, 8), 256, 0, stream>>>(x, xp);

    // Dense per-patch matrix M_s = B2_s * B1_s via 16 stages on identity
    bf_init_identity<<<KKPAT * NCH * NCH / 256, 256, 0, stream>>>(bufA);
    const int strides1[8] = {128, 64, 32, 16, 8, 4, 2, 1};   // increasing=False
    const int strides2[8] = {1, 2, 4, 8, 16, 32, 64, 128};   // increasing=True
    float* src = bufA;
    float* dst = bufB;
    for (int i = 0; i < 16; ++i) {
        int st = (i < 8) ? strides1[i] : strides2[i - 8];
        const float* tw = (i < 8) ? tw1 : tw2;
        bf_stage<<<KKPAT * 128 * NCH / 256, 256, 0, stream>>>(tw, st, src, dst);
        float* tmp = src; src = dst; dst = tmp;
    }
    bf_finalize<<<KDIM * NCH / 256, 256, 0, stream>>>(src, Wp);
    bf_bias<<<1, 256, 0, stream>>>(bias, bsum);

    dim3 grid(ROWS / 256, NCH / 64);
    butterfly_gemm_wmma<<<grid, 256, 0, stream>>>(xp, Wp, bsum, out);
}